// DFTBSK_9783935500642
// MI455X (gfx1250) — compile-verified
//
#include <hip/hip_runtime.h>
#include <hip/hip_bf16.h>

// ---------------- problem constants (from reference) ----------------
#define TPB        256
#define EPB        8192                 // edges per block
#define NGRID      512
#define NBOND      10
#define NING       16
#define TABLE_ELEMS (NBOND * NING * NGRID)   // 81920 floats
#define TABLE_BYTES (TABLE_ELEMS * 4)        // 327680 B == 320 KB (max LDS/WG)
#define X0F        1.0f
#define INV_DX     (511.0f / 9.0f)           // 1/DX, DX = (10-1)/(512-1)

typedef float v4f __attribute__((ext_vector_type(4)));
typedef __attribute__((address_space(1))) int g_i32;   // global b32
typedef __attribute__((address_space(3))) int l_i32;   // LDS b32

// ---------------- CDNA5 async global->LDS staging ----------------
#if defined(__gfx1250__) && __has_builtin(__builtin_amdgcn_global_load_async_to_lds_b32)
#define USE_ASYNC 1
#else
#define USE_ASYNC 0
#endif

__device__ __forceinline__ void wait_async_then_barrier() {
#if USE_ASYNC
#if __has_builtin(__builtin_amdgcn_s_wait_asynccnt)
  __builtin_amdgcn_s_wait_asynccnt(0);
#else
  asm volatile("s_wait_asynccnt 0" ::: "memory");
#endif
#endif
  __syncthreads();
}

// Stage one [B, M, G] f32 table into LDS transposed as [B, G, M].
// Global side: lane-consecutive linear indices -> fully coalesced 128B segments.
// LDS side: per-lane scattered b32 destinations perform the transpose in-flight.
__device__ __forceinline__ void stage_table(const float* __restrict__ src,
                                            float* smem) {
  const int tid = threadIdx.x;
#pragma unroll 4
  for (int i = tid; i < TABLE_ELEMS; i += TPB) {
    const int b   = i >> 13;        // / (16*512)
    const int rem = i & 8191;
    const int m   = rem >> 9;       // / 512
    const int g   = rem & 511;
    const int dst = (((b << 9) | g) << 4) | m;   // (b*512 + g)*16 + m
#if USE_ASYNC
    __builtin_amdgcn_global_load_async_to_lds_b32(
        (g_i32*)(src + i),
        (l_i32*)(smem + dst),
        0, 0);
#else
    smem[dst] = src[i];
#endif
  }
  wait_async_then_barrier();
}

// ---------------- per-edge interpolation from LDS ----------------
__device__ __forceinline__ void edge_pass(const float* smem,
                                          const float* __restrict__ rij,
                                          const int*   __restrict__ etype,
                                          float*       __restrict__ out,
                                          int e0, int e1) {
  for (int e = e0 + (int)threadIdx.x; e < e1; e += TPB) {
    const float r = rij[e];
    const int   b = etype[e];
    const float t = (r - X0F) * INV_DX;
    int i0 = (int)t;                              // t >= 0 here; trunc == floor
    i0 = i0 < 0 ? 0 : (i0 > NGRID - 2 ? NGRID - 2 : i0);
    const float frac = t - (float)i0;
    const float w0   = 1.0f - frac;

    // row pair: 2 x 64B contiguous -> 8 x ds_load_b128
    const v4f* row = (const v4f*)(smem + ((((b << 9) | i0) << 4)));
    const v4f y00 = row[0], y01 = row[1], y02 = row[2], y03 = row[3];
    const v4f y10 = row[4], y11 = row[5], y12 = row[6], y13 = row[7];

    v4f* op = (v4f*)(out + (size_t)e * NING);     // 64B-aligned, coalesced
    __builtin_nontemporal_store(y00 * w0 + y10 * frac, op + 0);
    __builtin_nontemporal_store(y01 * w0 + y11 * frac, op + 1);
    __builtin_nontemporal_store(y02 * w0 + y12 * frac, op + 2);
    __builtin_nontemporal_store(y03 * w0 + y13 * frac, op + 3);
  }
}

// ---------------- kernels ----------------
__global__ void __launch_bounds__(TPB, 1)
dftb_edges_kernel(const float* __restrict__ rij,
                  const int*   __restrict__ etype,
                  const float* __restrict__ hop,
                  const float* __restrict__ ovl,
                  float* __restrict__ out_ef,
                  float* __restrict__ out_eo,
                  int E) {
  extern __shared__ float smem[];                 // 320 KB dynamic LDS
  const int e0 = blockIdx.x * EPB;
  const int e1 = (E < e0 + EPB) ? E : (e0 + EPB);

  // Phase A: hopping table resident in LDS
  stage_table(hop, smem);
  edge_pass(smem, rij, etype, out_ef, e0, e1);
  __syncthreads();                                // all waves done reading LDS

  // Phase B: overlap table overwrites LDS
  stage_table(ovl, smem);
  edge_pass(smem, rij, etype, out_eo, e0, e1);
}

__global__ void dftb_nodes_kernel(const int* __restrict__ atype,
                                  const v4f* __restrict__ onsite,   // [4][4] f32
                                  v4f*       __restrict__ outn,
                                  int N) {
  const int n = blockIdx.x * blockDim.x + threadIdx.x;
  if (n < N) {
    const int a = atype[n];
    __builtin_nontemporal_store(onsite[a], outn + n);
  }
}

// ---------------- host entry ----------------
extern "C" void kernel_launch(void* const* d_in, const int* in_sizes, int n_in,
                              void* d_out, int out_size, void* d_ws, size_t ws_size,
                              hipStream_t stream) {
  const float* rij    = (const float*)d_in[0];
  const int*   etype  = (const int*)  d_in[1];
  const int*   atype  = (const int*)  d_in[2];
  const float* hop    = (const float*)d_in[3];
  const float* ovl    = (const float*)d_in[4];
  const float* onsite = (const float*)d_in[5];

  const int E = in_sizes[0];      // 2,000,000 edges
  const int N = in_sizes[2];      // 500,000 nodes

  float* out_ef = (float*)d_out;
  float* out_eo = out_ef + (size_t)E * NING;
  float* out_nd = out_eo + (size_t)E * NING;

  // Allow the full 320 KB WGP LDS allocation (idempotent; no stream work).
  (void)hipFuncSetAttribute(reinterpret_cast<const void*>(dftb_edges_kernel),
                            hipFuncAttributeMaxDynamicSharedMemorySize,
                            TABLE_BYTES);

  const int nblkE = (E + EPB - 1) / EPB;
  dftb_edges_kernel<<<nblkE, TPB, TABLE_BYTES, stream>>>(
      rij, etype, hop, ovl, out_ef, out_eo, E);

  const int nblkN = (N + TPB - 1) / TPB;
  dftb_nodes_kernel<<<nblkN, TPB, 0, stream>>>(
      atype, (const v4f*)onsite, (v4f*)out_nd, N);
}